// PointNetSetAbstraction_75969381532365
// MI455X (gfx1250) — compile-verified
//
#include <hip/hip_runtime.h>
#include <hip/hip_bf16.h>
#include <math.h>

// ---------------- problem constants ----------------
#define B_     16
#define N_     4096
#define DFEAT  20
#define S_     1024
#define K_     32
#define CIN_   24                       // 3 (xyz) + 1 (norm) + 20 (feat)
#define M_     (B_ * S_ * K_)           // 524288 columns for the MLP GEMMs
#define BN_EPS 1e-5f

typedef float v2f __attribute__((ext_vector_type(2)));
typedef float v8f __attribute__((ext_vector_type(8)));

// ---------------- workspace layout (float elements) ----------------
// Activations x0/x1 are stored channel-PAIR packed: float2 plane p holds channels (2p, 2p+1),
// linear index (p * M_ + m) in float2 units. total ~48.8M floats ~195MB.
constexpr size_t F_COMB  = 0;                                   // [B][24][N] channel-major combined pts (23 used)
constexpr size_t F_PNORM = F_COMB  + (size_t)B_ * 24 * N_;      // [B][N]
constexpr size_t F_QCOMB = F_PNORM + (size_t)B_ * N_;           // [B*S][24]
constexpr size_t F_QNORM = F_QCOMB + (size_t)B_ * S_ * 24;      // [B*S]
constexpr size_t F_X0    = F_QNORM + (size_t)B_ * S_;           // 12 float2-planes = [24][M]
constexpr size_t F_X1    = F_X0    + (size_t)24 * M_;           // 32 float2-planes = [64][M]
constexpr size_t F_STAT  = F_X1    + (size_t)64 * M_;           // 3 layers * 512 (sum|sumsq|scale|shift x128)
constexpr size_t F_END   = F_STAT  + 3 * 512;
constexpr size_t I_FPS   = F_END;                               // int region (element-indexed)
constexpr size_t I_KNN   = I_FPS + (size_t)B_ * S_;             // [B*S][K]

// d_out: new_xyz^T [B,3,S] | out [B,128,S] | new_bin [B,S,20]
constexpr size_t OUT_XYZ  = 0;
constexpr size_t OUT_FEAT = (size_t)B_ * 3 * S_;
constexpr size_t OUT_BIN  = OUT_FEAT + (size_t)B_ * 128 * S_;

// ---------------- init: zero stats + zero the atomic-max output region ----------------
__global__ void k_init(float* stats, float* out1) {
  int i = blockIdx.x * blockDim.x + threadIdx.x;
  if (i < 3 * 512) stats[i] = 0.f;
  if (i < B_ * 128 * S_) out1[i] = 0.f;   // ReLU >= 0, so 0 is the atomicMax identity
}

// ---------------- build combined points [B][24][N] + point norms ----------------
__global__ void k_build_comb(const float* __restrict__ xyz, const float* __restrict__ feat,
                             float* __restrict__ comb, float* __restrict__ pnorm) {
  int t = blockIdx.x * blockDim.x + threadIdx.x;
  if (t >= B_ * N_) return;
  int b = t / N_, n = t % N_;
  float* cb = comb + (size_t)b * 24 * N_;
  float acc = 0.f;
#pragma unroll
  for (int c = 0; c < 3; ++c) {
    float v = xyz[((size_t)b * 3 + c) * N_ + n];
    cb[(size_t)c * N_ + n] = v; acc += v * v;
  }
#pragma unroll
  for (int j = 0; j < DFEAT; ++j) {
    float v = feat[((size_t)b * N_ + n) * DFEAT + j];
    cb[(size_t)(3 + j) * N_ + n] = v; acc += v * v;
  }
  cb[(size_t)23 * N_ + n] = 0.f;
  pnorm[(size_t)b * N_ + n] = acc;
}

// ---------------- farthest point sampling: one workgroup per batch ----------------
__global__ void k_fps(const float* __restrict__ comb, int* __restrict__ fps_idx) {
  const int b = blockIdx.x, t = threadIdx.x;
  const float* cb = comb + (size_t)b * 24 * N_;
  __shared__ float cent[23];
  __shared__ float sval[256];
  __shared__ int   sidx[256];
  float dist[16];
#pragma unroll
  for (int j = 0; j < 16; ++j) dist[j] = 1e10f;
  int far = 0;
  for (int it = 0; it < S_; ++it) {
    if (t == 0) fps_idx[b * S_ + it] = far;
    if (t < 23) cent[t] = cb[(size_t)t * N_ + far];
    __syncthreads();
    float bv = -1.f; int bi = 0;
#pragma unroll
    for (int j = 0; j < 16; ++j) {
      int n = t + 256 * j;
      float d = 0.f;
      for (int c = 0; c < 23; ++c) { float df = cb[(size_t)c * N_ + n] - cent[c]; d += df * df; }
      float nd = fminf(dist[j], d); dist[j] = nd;
      if (nd > bv) { bv = nd; bi = n; }          // strict > keeps smallest index on ties
    }
    sval[t] = bv; sidx[t] = bi;
    __syncthreads();
    for (int s = 128; s > 0; s >>= 1) {
      if (t < s) {
        float ov = sval[t + s]; int oi = sidx[t + s];
        if (ov > sval[t] || (ov == sval[t] && oi < sidx[t])) { sval[t] = ov; sidx[t] = oi; }
      }
      __syncthreads();
    }
    far = sidx[0];
  }
}

// ---------------- gather queries; emit new_xyz^T and new_bin ----------------
__global__ void k_gather_q(const float* __restrict__ comb, const float* __restrict__ pnorm,
                           const int* __restrict__ fps_idx,
                           float* __restrict__ qcomb, float* __restrict__ qnorm,
                           float* __restrict__ out_xyz, float* __restrict__ out_bin) {
  int t = blockIdx.x * blockDim.x + threadIdx.x;
  if (t >= B_ * S_) return;
  int b = t / S_, s = t % S_;
  int idx = fps_idx[b * S_ + s];
  const float* cb = comb + (size_t)b * 24 * N_;
  float* q = qcomb + (size_t)t * 24;
  float acc = 0.f;
#pragma unroll
  for (int c = 0; c < 23; ++c) {
    float v = cb[(size_t)c * N_ + idx];
    q[c] = v; acc += v * v;
    if (c < 3) out_xyz[((size_t)b * 3 + c) * S_ + s] = v;
    else       out_bin[(size_t)t * DFEAT + (c - 3)] = v;
  }
  q[23] = 0.f;
  qnorm[t] = acc;
}

// ---------------- KNN: one workgroup per query; 32 min-selection rounds ----------------
__global__ void k_knn(const float* __restrict__ comb, const float* __restrict__ pnorm,
                      const float* __restrict__ qcomb, const float* __restrict__ qnorm,
                      int* __restrict__ knn_idx) {
  const int q = blockIdx.x;            // b*S + s
  const int b = q / S_;
  const int t = threadIdx.x;
  __shared__ float qv[24];
  __shared__ float dist[N_];           // 16KB
  __shared__ float sval[256];
  __shared__ int   sidx[256];
  if (t < 24) qv[t] = qcomb[(size_t)q * 24 + t];
  __syncthreads();
  const float qn = qnorm[q];
  const float* cb = comb + (size_t)b * 24 * N_;
  const float* pn = pnorm + (size_t)b * N_;
#pragma unroll 4
  for (int j = 0; j < 16; ++j) {
    int n = t + 256 * j;
    float dot = 0.f;
    for (int c = 0; c < 23; ++c) dot += qv[c] * cb[(size_t)c * N_ + n];
    dist[n] = qn + pn[n] - 2.f * dot;  // matches reference's -2*einsum + norms form
  }
  __syncthreads();
  for (int r = 0; r < K_; ++r) {
    float bv = 3.4e38f; int bi = 0x7fffffff;
#pragma unroll
    for (int j = 0; j < 16; ++j) {
      int n = t + 256 * j; float v = dist[n];
      if (v < bv || (v == bv && n < bi)) { bv = v; bi = n; }
    }
    sval[t] = bv; sidx[t] = bi;
    __syncthreads();
    for (int s = 128; s > 0; s >>= 1) {
      if (t < s) {
        float ov = sval[t + s]; int oi = sidx[t + s];
        if (ov < sval[t] || (ov == sval[t] && oi < sidx[t])) { sval[t] = ov; sidx[t] = oi; }
      }
      __syncthreads();
    }
    if (t == 0) { knn_idx[(size_t)q * K_ + r] = sidx[0]; dist[sidx[0]] = 3.4e38f; }
    __syncthreads();
  }
}

// ---------------- build x0 (channel-pair packed): gxyz_norm(3) | L2 norm(1) | gdep_norm(20) ----------------
__global__ void k_build_x0(const float* __restrict__ comb, const float* __restrict__ qcomb,
                           const int* __restrict__ knn_idx, float* __restrict__ x0) {
  int m = blockIdx.x * blockDim.x + threadIdx.x;
  if (m >= M_) return;
  int k = m % K_;
  int sg = m / K_;           // b*S + s
  int b = sg / S_;
  int n = knn_idx[(size_t)sg * K_ + k];
  const float* cb = comb + (size_t)b * 24 * N_;
  const float* qv = qcomb + (size_t)sg * 24;
  float vals[24];
#pragma unroll
  for (int c = 0; c < 3; ++c) vals[c] = cb[(size_t)c * N_ + n] - qv[c];
  vals[3] = sqrtf(fmaxf(vals[0] * vals[0] + vals[1] * vals[1] + vals[2] * vals[2], 1e-12f));
#pragma unroll
  for (int j = 0; j < DFEAT; ++j) vals[4 + j] = cb[(size_t)(3 + j) * N_ + n] - qv[3 + j];
  v2f* x0p = (v2f*)x0;
#pragma unroll
  for (int p = 0; p < 12; ++p) {
    v2f t2; t2.x = vals[2 * p]; t2.y = vals[2 * p + 1];
    x0p[(size_t)p * M_ + m] = t2;
  }
}

// ---------------- fused GEMM (+bias via WMMA C-init from LDS) using V_WMMA_F32_16X16X4_F32 ----------------
// Weights + bias staged in LDS; activations channel-pair packed so each B-fragment is one b64 load.
// MODE 0: accumulate per-channel sum / sumsq (BN training stats)
// MODE 1: relu(y*scale + shift), packed store (in-place safe: full 64-channel column per wave)
// Each wave: 64 output channels (4 o-tiles) x MT=8 column tiles of 16.
template <int CI, int MODE>
__global__ void k_gemm(const float* xin, const float* __restrict__ w,
                       const float* __restrict__ bias,
                       float* __restrict__ gsum, float* __restrict__ gsq,
                       const float* __restrict__ scale, const float* __restrict__ shift,
                       float* xout) {
  constexpr int OT = 4, MT = 8, KS = CI / 4;
  __shared__ __attribute__((aligned(32))) float wlds[64 * CI];
  __shared__ __attribute__((aligned(32))) float blds[64];   // bias -> WMMA C initializer
  __shared__ float lsum[64];
  __shared__ float lsq[64];
#pragma unroll 1
  for (int i = threadIdx.x; i < 64 * CI; i += 256) wlds[i] = w[i];
  if (threadIdx.x < 64) {
    blds[threadIdx.x] = bias[threadIdx.x];
    if (MODE == 0) { lsum[threadIdx.x] = 0.f; lsq[threadIdx.x] = 0.f; }
  }
  __syncthreads();

  const int lane = threadIdx.x & 31;
  const int wv   = threadIdx.x >> 5;
  const int half = lane >> 4;
  const int l    = lane & 15;
  const unsigned mstart = ((unsigned)blockIdx.x * 8 + wv) * (MT * 16);

  const v2f* xin2 = (const v2f*)xin;
  v2f* xout2 = (v2f*)xout;

  // per-lane BN constants (MODE 1 only): channel = ot*16 + v + 8*half
  float e0[OT][8], e1[OT][8];
  if (MODE == 1) {
#pragma unroll
    for (int ot = 0; ot < OT; ++ot)
#pragma unroll
      for (int v = 0; v < 8; ++v) {
        int ch = ot * 16 + v + 8 * half;
        e0[ot][v] = scale[ch]; e1[ot][v] = shift[ch];
      }
  }

  const v8f vzero = {0.f, 0.f, 0.f, 0.f, 0.f, 0.f, 0.f, 0.f};
  v8f sumac[OT], sqac[OT];
  if (MODE == 0)
#pragma unroll
    for (int ot = 0; ot < OT; ++ot) { sumac[ot] = vzero; sqac[ot] = vzero; }

#pragma unroll 1
  for (int mt = 0; mt < MT; ++mt) {
    const unsigned m0l = mstart + (unsigned)mt * 16 + l;
    v8f acc[OT];
#pragma unroll
    for (int ot = 0; ot < OT; ++ot)                          // y = Wx + bias: C-init from LDS
      acc[ot] = *(const v8f*)&blds[ot * 16 + 8 * half];      // 32B-aligned ds_load
#pragma unroll
    for (int ks = 0; ks < KS; ++ks) {
      const int kb = ks * 4 + half * 2;                      // k-rows handled by this half
      const v2f bf = xin2[(unsigned)(ks * 2 + half) * M_ + m0l];  // rows kb,kb+1 col l: one b64
#pragma unroll
      for (int ot = 0; ot < OT; ++ot) {
        const v2f af = *(const v2f*)&wlds[(ot * 16 + l) * CI + kb]; // ds_load_b64
        acc[ot] = __builtin_amdgcn_wmma_f32_16x16x4_f32(
            false, af, false, bf, (short)0, acc[ot], false, false);
      }
    }
    if (MODE == 0) {
#pragma unroll
      for (int ot = 0; ot < OT; ++ot) { sumac[ot] += acc[ot]; sqac[ot] += acc[ot] * acc[ot]; }
    } else {
#pragma unroll
      for (int ot = 0; ot < OT; ++ot)
#pragma unroll
        for (int j = 0; j < 4; ++j) {                        // channel pair (2j,2j+1) of this half
          v2f o2;
          o2.x = fmaxf(acc[ot][2 * j]     * e0[ot][2 * j]     + e1[ot][2 * j],     0.f);
          o2.y = fmaxf(acc[ot][2 * j + 1] * e0[ot][2 * j + 1] + e1[ot][2 * j + 1], 0.f);
          xout2[(unsigned)(ot * 8 + j + 4 * half) * M_ + m0l] = o2;
        }
    }
    __syncthreads();   // pins LDS A-frag/bias loads inside the loop (prevents hoist/spill)
  }

  if (MODE == 0) {
#pragma unroll
    for (int ot = 0; ot < OT; ++ot)
#pragma unroll
      for (int v = 0; v < 8; ++v) {
        float sv = sumac[ot][v], q2 = sqac[ot][v];
#pragma unroll
        for (int d = 1; d < 16; d <<= 1) {
          sv += __shfl_xor(sv, d, 16);
          q2 += __shfl_xor(q2, d, 16);
        }
        if (l == 0) {
          int ch = ot * 16 + v + 8 * half;
          atomicAdd(&lsum[ch], sv);
          atomicAdd(&lsq[ch], q2);
        }
      }
    __syncthreads();
    if (threadIdx.x < 64) {
      atomicAdd(&gsum[threadIdx.x], lsum[threadIdx.x]);
      atomicAdd(&gsq[threadIdx.x], lsq[threadIdx.x]);
    }
  }
}

// ---------------- BN finalize: fold (mu, var, gamma, beta) -> scale/shift ----------------
__global__ void k_bnfin(const float* __restrict__ gsum, const float* __restrict__ gsq,
                        const float* __restrict__ gamma, const float* __restrict__ beta,
                        float* __restrict__ scale, float* __restrict__ shift, int O) {
  int o = blockIdx.x * blockDim.x + threadIdx.x;
  if (o >= O) return;
  const float inv_n = 1.f / (float)M_;
  float mu  = gsum[o] * inv_n;
  float var = gsq[o] * inv_n - mu * mu;      // biased var, matches reference
  float a = gamma[o] * rsqrtf(var + BN_EPS);
  scale[o] = a;
  shift[o] = beta[o] - mu * a;
}

// ---------------- final layer: GEMM + BN + ReLU + max-over-K via monotone uint atomicMax ----------------
__global__ void k_gemm_max(const float* __restrict__ xin, const float* __restrict__ w,
                           const float* __restrict__ bias, const float* __restrict__ scale,
                           const float* __restrict__ shift, float* __restrict__ outp) {
  constexpr int OT = 4, MT = 8, CI = 64, KS = CI / 4;
  __shared__ __attribute__((aligned(32))) float wlds[64 * CI];
  __shared__ __attribute__((aligned(32))) float blds[64];
#pragma unroll 1
  for (int i = threadIdx.x; i < 64 * CI; i += 256) wlds[i] = w[i];
  if (threadIdx.x < 64) blds[threadIdx.x] = bias[threadIdx.x];
  __syncthreads();

  const int lane = threadIdx.x & 31;
  const int wv   = threadIdx.x >> 5;
  const int half = lane >> 4;
  const int l    = lane & 15;
  const unsigned mstart = ((unsigned)blockIdx.x * 8 + wv) * (MT * 16);
  const v2f* xin2 = (const v2f*)xin;

  float e0[OT][8], e1[OT][8];
#pragma unroll
  for (int ot = 0; ot < OT; ++ot)
#pragma unroll
    for (int v = 0; v < 8; ++v) {
      int ch = ot * 16 + v + 8 * half;
      e0[ot][v] = scale[ch];
      e1[ot][v] = shift[ch];
    }

#pragma unroll 1
  for (int mt = 0; mt < MT; ++mt) {
    const unsigned m0 = mstart + (unsigned)mt * 16;   // 16 columns, all same (b,s) since K_=32
    const unsigned m0l = m0 + l;
    v8f acc[OT];
#pragma unroll
    for (int ot = 0; ot < OT; ++ot)
      acc[ot] = *(const v8f*)&blds[ot * 16 + 8 * half];
#pragma unroll
    for (int ks = 0; ks < KS; ++ks) {
      const int kb = ks * 4 + half * 2;
      const v2f bf = xin2[(unsigned)(ks * 2 + half) * M_ + m0l];
#pragma unroll
      for (int ot = 0; ot < OT; ++ot) {
        const v2f af = *(const v2f*)&wlds[(ot * 16 + l) * CI + kb];
        acc[ot] = __builtin_amdgcn_wmma_f32_16x16x4_f32(
            false, af, false, bf, (short)0, acc[ot], false, false);
      }
    }
    const unsigned sg = m0 / K_;                      // b*S + s
    const int b = (int)(sg / S_);
    const int s = (int)(sg % S_);
#pragma unroll
    for (int ot = 0; ot < OT; ++ot)
#pragma unroll
      for (int v = 0; v < 8; ++v) {
        float val = fmaxf(acc[ot][v] * e0[ot][v] + e1[ot][v], 0.f);
#pragma unroll
        for (int d = 1; d < 16; d <<= 1) val = fmaxf(val, __shfl_xor(val, d, 16));
        if (l == 0) {
          int ch = ot * 16 + v + 8 * half;
          unsigned int* addr = (unsigned int*)&outp[((size_t)b * 128 + ch) * S_ + s];
          atomicMax(addr, __float_as_uint(val));      // valid: val >= 0, init 0
        }
      }
    __syncthreads();
  }
}

// ---------------- host launch ----------------
extern "C" void kernel_launch(void* const* d_in, const int* in_sizes, int n_in,
                              void* d_out, int out_size, void* d_ws, size_t ws_size,
                              hipStream_t stream) {
  (void)in_sizes; (void)n_in; (void)out_size; (void)ws_size;
  const float* xyz  = (const float*)d_in[0];
  const float* feat = (const float*)d_in[1];
  const float* w0 = (const float*)d_in[2];  const float* b0 = (const float*)d_in[3];
  const float* g0 = (const float*)d_in[4];  const float* bt0 = (const float*)d_in[5];
  const float* w1 = (const float*)d_in[6];  const float* b1 = (const float*)d_in[7];
  const float* g1 = (const float*)d_in[8];  const float* bt1 = (const float*)d_in[9];
  const float* w2 = (const float*)d_in[10]; const float* b2 = (const float*)d_in[11];
  const float* g2 = (const float*)d_in[12]; const float* bt2 = (const float*)d_in[13];

  float* ws     = (float*)d_ws;
  float* comb   = ws + F_COMB;
  float* pnorm  = ws + F_PNORM;
  float* qcomb  = ws + F_QCOMB;
  float* qnorm  = ws + F_QNORM;
  float* x0     = ws + F_X0;
  float* x1     = ws + F_X1;
  float* stats  = ws + F_STAT;          // per layer L: +L*512 -> sum[128]|sumsq[128]|scale[128]|shift[128]
  int*   fpsix  = (int*)(ws + I_FPS);
  int*   knnix  = (int*)(ws + I_KNN);

  float* out    = (float*)d_out;
  float* o_xyz  = out + OUT_XYZ;
  float* o_feat = out + OUT_FEAT;
  float* o_bin  = out + OUT_BIN;

  // stage 0: init
  k_init<<<(B_ * 128 * S_ + 255) / 256, 256, 0, stream>>>(stats, o_feat);
  // stage 1: combined points + norms
  k_build_comb<<<(B_ * N_ + 255) / 256, 256, 0, stream>>>(xyz, feat, comb, pnorm);
  // stage 2: FPS (one WG per batch)
  k_fps<<<B_, 256, 0, stream>>>(comb, fpsix);
  // stage 3: gather queries, emit new_xyz^T and new_bin
  k_gather_q<<<(B_ * S_ + 255) / 256, 256, 0, stream>>>(comb, pnorm, fpsix, qcomb, qnorm, o_xyz, o_bin);
  // stage 4: KNN top-32 per query
  k_knn<<<B_ * S_, 256, 0, stream>>>(comb, pnorm, qcomb, qnorm, knnix);
  // stage 5: grouped/normalized input x0 (packed)
  k_build_x0<<<(M_ + 255) / 256, 256, 0, stream>>>(comb, qcomb, knnix, x0);

  const int GB = M_ / (16 * 8 * 8);     // 512 blocks: 8 waves/block, 8 m-tiles/wave
  float* st0 = stats + 0 * 512; float* st1 = stats + 1 * 512; float* st2 = stats + 2 * 512;

  // layer 0: 24 -> 64
  k_gemm<24, 0><<<GB, 256, 0, stream>>>(x0, w0, b0, st0, st0 + 128, nullptr, nullptr, nullptr);
  k_bnfin<<<1, 64, 0, stream>>>(st0, st0 + 128, g0, bt0, st0 + 256, st0 + 384, 64);
  k_gemm<24, 1><<<GB, 256, 0, stream>>>(x0, w0, b0, nullptr, nullptr, st0 + 256, st0 + 384, x1);

  // layer 1: 64 -> 64 (pass 2 in-place: each wave does the full 64-channel column)
  k_gemm<64, 0><<<GB, 256, 0, stream>>>(x1, w1, b1, st1, st1 + 128, nullptr, nullptr, nullptr);
  k_bnfin<<<1, 64, 0, stream>>>(st1, st1 + 128, g1, bt1, st1 + 256, st1 + 384, 64);
  k_gemm<64, 1><<<GB, 256, 0, stream>>>(x1, w1, b1, nullptr, nullptr, st1 + 256, st1 + 384, x1);

  // layer 2: 64 -> 128, two 64-channel halves; stats then fused BN+ReLU+max-over-K
  k_gemm<64, 0><<<GB, 256, 0, stream>>>(x1, w2,           b2,      st2,      st2 + 128, nullptr, nullptr, nullptr);
  k_gemm<64, 0><<<GB, 256, 0, stream>>>(x1, w2 + 64 * 64, b2 + 64, st2 + 64, st2 + 128 + 64, nullptr, nullptr, nullptr);
  k_bnfin<<<1, 128, 0, stream>>>(st2, st2 + 128, g2, bt2, st2 + 256, st2 + 384, 128);
  k_gemm_max<<<GB, 256, 0, stream>>>(x1, w2,           b2,      st2 + 256,      st2 + 384,      o_feat);
  k_gemm_max<<<GB, 256, 0, stream>>>(x1, w2 + 64 * 64, b2 + 64, st2 + 256 + 64, st2 + 384 + 64, o_feat + (size_t)64 * S_);
}